// LocationSensitiveAttention_22196390986443
// MI455X (gfx1250) — compile-verified
//
#include <hip/hip_runtime.h>
#include <hip/hip_bf16.h>
#include <math.h>

// Problem constants (match reference)
#define T_   2048
#define B_   256
#define H_   256
#define Q_   1024
#define Kc_  31
#define WL_  256
#define PAD_ 15   // (Kc_-1)/2

typedef __attribute__((ext_vector_type(2))) float v2f;
typedef __attribute__((ext_vector_type(8))) float v8f;

// ---------------------------------------------------------------------------
// Kernel 1: q = query @ wq^T + bq   ([B,Q] x [H,Q]^T -> [B,H]), fp32 WMMA.
// One wave per 16x16 output tile. K=1024 consumed 4 at a time with
// V_WMMA_F32_16X16X4_F32 (fp32 in/out -> bit-compatible precision with ref).
//
// Fragment layouts (ISA 7.12.2):
//   A (16x4, MxK):  comp r of lane l -> M = l%16, K = 2*(l>=16) + r
//   B (4x16, KxN):  comp r of lane l -> N = l%16, K = 2*(l>=16) + r
//   C/D (16x16):    comp r of lane l -> M = r + 8*(l>=16), N = l%16
// ---------------------------------------------------------------------------
__global__ __launch_bounds__(32) void qproj_wmma_kernel(
    const float* __restrict__ query,  // [B_, Q_]
    const float* __restrict__ wq,     // [H_, Q_] (row-major; B = wq^T)
    const float* __restrict__ bq,     // [H_]
    float* __restrict__ qout)         // [B_, H_]
{
    const int n0   = blockIdx.x * 16;   // H tile
    const int m0   = blockIdx.y * 16;   // B tile
    const int lane = threadIdx.x;       // 0..31, wave32
    const int half = lane >> 4;         // 0 | 1
    const int lm   = lane & 15;

    const float* arow = query + (size_t)(m0 + lm) * Q_ + 2 * half;
    const float* brow = wq    + (size_t)(n0 + lm) * Q_ + 2 * half;

    v8f acc = {};
    for (int kb = 0; kb < Q_; kb += 4) {
        v2f a, bm;
        a.x  = arow[kb + 0];
        a.y  = arow[kb + 1];
        bm.x = brow[kb + 0];
        bm.y = brow[kb + 1];
        // 8 args: (neg_a, A, neg_b, B, c_mod, C, reuse_a, reuse_b)
        acc = __builtin_amdgcn_wmma_f32_16x16x4_f32(
            false, a, false, bm, (short)0, acc, false, false);
    }

    const int n = n0 + lm;
    const float bias = bq[n];
#pragma unroll
    for (int r = 0; r < 8; ++r) {
        const int m = m0 + r + 8 * half;
        qout[(size_t)m * H_ + n] = acc[r] + bias;
    }
}

// ---------------------------------------------------------------------------
// Kernel 2: everything else. One workgroup (256 threads = 8 waves) per batch
// element. LDS stages the 286-float location window, the conv filter
// (256x31), q-row, w_score, conv bias. Phase 1: thread t computes score[t]
// (conv + tanh + dot). Phase 2: LDS softmax. Phase 3: thread h accumulates
// context[h] over the window with coalesced 1KB row reads of encoded_tokens.
// Phase 4: scatter alignment / cumulative over T; argmax -> new_start.
// ---------------------------------------------------------------------------
__global__ __launch_bounds__(256) void lsa_main_kernel(
    const float*         __restrict__ enc,      // [T_, B_, H_]
    const unsigned char* __restrict__ mask,     // [B_, T_] (bool)
    const float*         __restrict__ ca_in,    // [B_, T_]
    const float*         __restrict__ init_ca,  // [B_, 1]
    const int*           __restrict__ wstart,   // [B_]
    const float*         __restrict__ conv_w,   // [H_, 1, Kc_]
    const float*         __restrict__ conv_b,   // [H_]
    const float*         __restrict__ w_score,  // [H_]
    const float*         __restrict__ qproj,    // [B_, H_] (from kernel 1)
    float* __restrict__ out_ctx,                // [B_, H_]
    float* __restrict__ out_cum,                // [B_, T_]
    float* __restrict__ out_align,              // [B_, T_]
    float* __restrict__ out_ns)                 // [B_]
{
    const int b   = blockIdx.x;
    const int tid = threadIdx.x;

    __shared__ float lf[WL_ + 2 * PAD_];   // windowed location features
    __shared__ float qv[H_];
    __shared__ float wsc[H_];
    __shared__ float cb[H_];
    __shared__ float cw[H_][32];           // 31 taps padded to 32
    __shared__ float red[256];
    __shared__ int   redi[256];
    __shared__ float alignv[WL_];

    const int   ws   = wstart[b];
    const float icav = init_ca[b];

    // ---- stage shared data ----
    qv[tid]  = qproj[(size_t)b * H_ + tid];
    wsc[tid] = w_score[tid];
    cb[tid]  = conv_b[tid];
    for (int i = tid; i < H_ * Kc_; i += 256)
        cw[i / Kc_][i % Kc_] = conv_w[i];

    // lf (full signal) = [PAD x init | masked ca (T) | PAD x 0]; take window
    for (int j = tid; j < WL_ + 2 * PAD_; j += 256) {
        const int i = ws + j;          // index into full lf, always >= 0
        float v;
        if (i < PAD_) {
            v = icav;
        } else {
            const int t = i - PAD_;
            v = (t < T_) ? (mask[(size_t)b * T_ + t] ? ca_in[(size_t)b * T_ + t] : 0.0f)
                         : 0.0f;
        }
        lf[j] = v;
    }
    __syncthreads();

    // ---- phase 1: score[t] for t = tid ----
    float sc = 0.0f;
    {
        const int t = tid;
        for (int h = 0; h < H_; ++h) {
            float acc = cb[h];
#pragma unroll
            for (int k = 0; k < Kc_; ++k)
                acc = fmaf(cw[h][k], lf[t + k], acc);
            sc = fmaf(wsc[h], tanhf(acc + qv[h]), sc);
        }
    }
    const bool  mw  = mask[(size_t)b * T_ + ws + tid] != 0;
    const float scm = mw ? sc : -INFINITY;

    // ---- phase 2: softmax across the 256-wide window ----
    red[tid] = scm;
    __syncthreads();
    for (int s = 128; s > 0; s >>= 1) {
        if (tid < s) red[tid] = fmaxf(red[tid], red[tid + s]);
        __syncthreads();
    }
    const float mx = red[0];
    __syncthreads();
    const float ex = (scm == -INFINITY) ? 0.0f : expf(scm - mx);
    red[tid] = ex;
    __syncthreads();
    for (int s = 128; s > 0; s >>= 1) {
        if (tid < s) red[tid] += red[tid + s];
        __syncthreads();
    }
    const float denom = red[0];
    __syncthreads();
    const float al = ex / denom;
    alignv[tid] = al;
    __syncthreads();

    // ---- phase 3: context[h] = sum_t align[t] * enc[ws+t, b, h] ----
    {
        float ctx = 0.0f;
        const float* encb = enc + (size_t)b * H_ + tid;
        for (int t = 0; t < WL_; ++t)
            ctx = fmaf(alignv[t], encb[(size_t)(ws + t) * (B_ * H_)], ctx);
        out_ctx[(size_t)b * H_ + tid] = ctx;
    }

    // ---- phase 4: scatter alignment, cumulative ----
    for (int i = tid; i < T_; i += 256) {
        const float cam = mask[(size_t)b * T_ + i] ? ca_in[(size_t)b * T_ + i] : 0.0f;
        const int   rel = i - ws;
        const float av  = (rel >= 0 && rel < WL_) ? alignv[rel] : 0.0f;
        out_align[(size_t)b * T_ + i] = av;
        out_cum[(size_t)b * T_ + i]   = cam + av;
    }

    // ---- argmax over full-length alignment (zeros outside window) ----
    red[tid]  = al;
    redi[tid] = ws + tid;
    __syncthreads();
    for (int s = 128; s > 0; s >>= 1) {
        if (tid < s) {
            const float v2 = red[tid + s];
            const int   i2 = redi[tid + s];
            if (v2 > red[tid] || (v2 == red[tid] && i2 < redi[tid])) {
                red[tid]  = v2;
                redi[tid] = i2;
            }
        }
        __syncthreads();
    }
    if (tid == 0) {
        // if the window max is 0, the first global max is index 0 (all zeros)
        int idx = (red[0] > 0.0f) ? redi[0] : 0;
        int ns  = idx - WL_ / 2;
        ns = ns < 0 ? 0 : (ns > T_ - WL_ ? T_ - WL_ : ns);
        out_ns[b] = (float)ns;
    }
}

// ---------------------------------------------------------------------------
// Launch. Inputs in setup_inputs() order.
// ---------------------------------------------------------------------------
extern "C" void kernel_launch(void* const* d_in, const int* in_sizes, int n_in,
                              void* d_out, int out_size, void* d_ws, size_t ws_size,
                              hipStream_t stream) {
    const float*         enc     = (const float*)d_in[0];          // [T,B,H]
    const unsigned char* mask    = (const unsigned char*)d_in[1];  // [B,T] bool
    const float*         query   = (const float*)d_in[2];          // [1,B,Q]
    const float*         ca      = (const float*)d_in[3];          // [B,T]
    const float*         init_ca = (const float*)d_in[4];          // [B,1]
    const int*           wstart  = (const int*)d_in[5];            // [B]
    const float*         conv_w  = (const float*)d_in[6];          // [H,1,K]
    const float*         conv_b  = (const float*)d_in[7];          // [H]
    const float*         wq      = (const float*)d_in[8];          // [H,Q]
    const float*         bq      = (const float*)d_in[9];          // [H]
    const float*         w_score = (const float*)d_in[10];         // [H]

    float* out       = (float*)d_out;
    float* out_ctx   = out;                         // [B,H]
    float* out_cum   = out_ctx + B_ * H_;           // [B,T]
    float* out_align = out_cum + (size_t)B_ * T_;   // [B,T]
    float* out_ns    = out_align + (size_t)B_ * T_; // [B]

    float* qproj = (float*)d_ws;                    // [B,H] scratch

    // GEMM: one wave per 16x16 tile, 16x16 tiles over [B=256, H=256]
    qproj_wmma_kernel<<<dim3(H_ / 16, B_ / 16), 32, 0, stream>>>(query, wq, bq, qproj);

    // Rest: one workgroup per batch element
    lsa_main_kernel<<<B_, 256, 0, stream>>>(enc, mask, ca, init_ca, wstart,
                                            conv_w, conv_b, w_score, qproj,
                                            out_ctx, out_cum, out_align, out_ns);
}